// MultiHeadGraphAttentionNetwork_89627377533187
// MI455X (gfx1250) — compile-verified
//
#include <hip/hip_runtime.h>
#include <hip/hip_bf16.h>

#define NN    3072
#define HID   512
#define NH    8
#define HD    64
#define NE    98304
#define PCOLS 2560            // 512 (Wv) + 1024 (Ww top) + 1024 (Ww bot)
#define LN_EPS 1e-5f

typedef __attribute__((ext_vector_type(2))) float v2f;
typedef __attribute__((ext_vector_type(8))) float v8f;

// ---------------- workspace layout (4-byte units) ----------------
#define OFF_P      0u               // 3072*2560       = 7,864,320 f32
#define OFF_W      7864320u         // NH*NE           =   786,432 f32
#define OFF_PEXP   8650752u         // NH*NE           =   786,432 f32
#define OFF_RMK    9437184u         // NH*NN           =    24,576 u32 (max keys)
#define OFF_RSUM   9461760u         // NH*NN           =    24,576 f32
#define OFF_CNT    9486336u         // NN              =     3,072 i32
#define OFF_START  9489408u         // NN+1 (pad 3088) i32
#define OFF_CUR    9492496u         // NN              i32
#define OFF_CSR    9495568u         // NE              i32
#define ZERO_WORDS (24576u + 24576u + 3072u)   // rmk + rsum + counts, contiguous

// ---------------- init: zero rmk/rowsum/counts ----------------
__global__ __launch_bounds__(256) void init_kernel(unsigned* base) {
    unsigned t = blockIdx.x * 256u + threadIdx.x;
    if (t < ZERO_WORDS) base[t] = 0u;
}

// ---------------- fused GEMM: P = x @ [Wv | Ww_top | Ww_bot] + [bv | bw | 0] ----------------
// grid: (24, 20), block 256 = 8 waves. Workgroup tile 128(M) x 128(N), K-slabs of 16.
// Wave (mw,nw) = 32x64 strip -> 2x4 accumulators of 16x16, fp32 WMMA (16x16x4).
// B stored transposed in LDS so every WMMA operand is one aligned ds_load_b64.
__global__ __launch_bounds__(256) void gemm_kernel(const float* __restrict__ x,
                                                   const float* __restrict__ Wv,
                                                   const float* __restrict__ bv,
                                                   const float* __restrict__ Ww,
                                                   const float* __restrict__ bw,
                                                   float* __restrict__ P) {
    __shared__ float As[128][18];    // row-major, even stride -> 8B-aligned K pairs
    __shared__ float BsT[128][18];   // BsT[col][k], transposed

    const int m0 = blockIdx.x * 128;
    const int c0 = blockIdx.y * 128;

    const float* bsrc; int bstride; int bc0;
    if (c0 < 512)       { bsrc = Wv;              bstride = 512;  bc0 = c0;        }
    else if (c0 < 1536) { bsrc = Ww;              bstride = 1024; bc0 = c0 - 512;  }
    else                { bsrc = Ww + 512 * 1024; bstride = 1024; bc0 = c0 - 1536; }

    const int tid  = threadIdx.x;
    const int lane = tid & 31;
    const int wv   = tid >> 5;
    const int mw   = wv & 3;        // 0..3 : 32-row strip
    const int nw   = wv >> 2;       // 0..1 : 64-col half
    const int lrow = lane & 15;
    const int lhi  = lane >> 4;

    // loader mapping: A tile 128x16 (8 floats/thread), B tile 16x128 (8 floats/thread)
    const int ar = tid >> 1;              // A row 0..127
    const int ac = (tid & 1) * 8;         // A col 0 or 8
    const int br = tid >> 4;              // B k-row 0..15
    const int bc = (tid & 15) * 8;        // B col 0..120

    v8f acc[2][4];
    #pragma unroll
    for (int mf = 0; mf < 2; ++mf)
        #pragma unroll
        for (int nf = 0; nf < 4; ++nf) acc[mf][nf] = (v8f){0,0,0,0,0,0,0,0};

    for (int kt = 0; kt < HID / 16; ++kt) {
        const int k0 = kt * 16;
        // stage A (row-major)
        const float* ap = x + (m0 + ar) * HID + k0 + ac;
        const float4 a0 = *(const float4*)(ap);
        const float4 a1 = *(const float4*)(ap + 4);
        As[ar][ac + 0] = a0.x; As[ar][ac + 1] = a0.y; As[ar][ac + 2] = a0.z; As[ar][ac + 3] = a0.w;
        As[ar][ac + 4] = a1.x; As[ar][ac + 5] = a1.y; As[ar][ac + 6] = a1.z; As[ar][ac + 7] = a1.w;
        // stage B (transposed)
        const float* bp = bsrc + (k0 + br) * bstride + bc0 + bc;
        const float4 b0 = *(const float4*)(bp);
        const float4 b1 = *(const float4*)(bp + 4);
        BsT[bc + 0][br] = b0.x; BsT[bc + 1][br] = b0.y; BsT[bc + 2][br] = b0.z; BsT[bc + 3][br] = b0.w;
        BsT[bc + 4][br] = b1.x; BsT[bc + 5][br] = b1.y; BsT[bc + 6][br] = b1.z; BsT[bc + 7][br] = b1.w;
        if (kt + 1 < HID / 16) {   // prefetch next slab (global_prefetch_b8)
            __builtin_prefetch(ap + 16, 0, 3);
            __builtin_prefetch(bp + 16 * bstride, 0, 3);
        }
        __syncthreads();

        #pragma unroll
        for (int ks = 0; ks < 4; ++ks) {
            const int kk = 4 * ks + 2 * lhi;       // always even -> 8B aligned
            const v2f av0 = *(const v2f*)&As[mw * 32 + lrow][kk];
            const v2f av1 = *(const v2f*)&As[mw * 32 + 16 + lrow][kk];
            #pragma unroll
            for (int nf = 0; nf < 4; ++nf) {
                const int nc = nw * 64 + nf * 16 + lrow;
                const v2f bv2 = *(const v2f*)&BsT[nc][kk];
                acc[0][nf] = __builtin_amdgcn_wmma_f32_16x16x4_f32(
                    false, av0, false, bv2, (short)0, acc[0][nf], false, false);
                acc[1][nf] = __builtin_amdgcn_wmma_f32_16x16x4_f32(
                    false, av1, false, bv2, (short)0, acc[1][nf], false, false);
            }
        }
        __syncthreads();
    }

    // epilogue: bias + store. D layout: vgpr v -> M = v + 8*lhi, N = lrow
    #pragma unroll
    for (int nf = 0; nf < 4; ++nf) {
        const int cg = c0 + nw * 64 + nf * 16 + lrow;          // global column
        float bias = 0.f;
        if (cg < 512)       bias = bv[cg];
        else if (cg < 1536) bias = bw[cg - 512];
        #pragma unroll
        for (int mf = 0; mf < 2; ++mf) {
            #pragma unroll
            for (int v = 0; v < 8; ++v) {
                const int row = m0 + mw * 32 + mf * 16 + v + 8 * lhi;
                P[row * PCOLS + cg] = acc[mf][nf][v] + bias;
            }
        }
    }
}

// ---------------- edge scores: w[h,e] = ba + sum_k LR(P1[src]+P2[dst]) * Wa[k] ----------------
// one wave per (edge, head); also: atomic row-max (order-independent) + degree counts
__global__ __launch_bounds__(256) void edge_w_kernel(const float* __restrict__ P,
                                                     const int* __restrict__ edges,
                                                     const float* __restrict__ Wa,
                                                     const float* __restrict__ ba,
                                                     float* __restrict__ wbuf,
                                                     unsigned* __restrict__ rmk,
                                                     int* __restrict__ counts) {
    const int lane = threadIdx.x & 31;
    const int gid  = blockIdx.x * 8 + (threadIdx.x >> 5);  // 0 .. NH*NE-1
    const int e = gid >> 3;
    const int h = gid & 7;
    const int src = edges[e];
    const int dst = edges[NE + e];
    const float* r1 = P + src * PCOLS + 512  + h * 128;    // includes +bw
    const float* r2 = P + dst * PCOLS + 1536 + h * 128;
    float acc = 0.f;
    #pragma unroll
    for (int q = 0; q < 4; ++q) {
        const int j = lane + 32 * q;
        float v = r1[j] + r2[j];
        v = v > 0.f ? v : 0.01f * v;                       // leaky_relu(0.01)
        acc = fmaf(v, Wa[j], acc);
    }
    #pragma unroll
    for (int o = 16; o > 0; o >>= 1) acc += __shfl_xor(acc, o, 32);
    if (lane == 0) {
        const float wval = acc + ba[0];
        wbuf[h * NE + e] = wval;
        const unsigned b = __float_as_uint(wval);
        const unsigned key = (b & 0x80000000u) ? ~b : (b | 0x80000000u);  // order-preserving
        atomicMax(&rmk[h * NN + src], key);
        if (h == 0) atomicAdd(&counts[src], 1);
    }
}

// ---------------- serial prefix scan over 3072 row counts ----------------
__global__ void scan_kernel(const int* __restrict__ counts,
                            int* __restrict__ rowstart, int* __restrict__ cursor) {
    if (threadIdx.x == 0 && blockIdx.x == 0) {
        int s = 0;
        for (int i = 0; i < NN; ++i) { rowstart[i] = s; cursor[i] = s; s += counts[i]; }
        rowstart[NN] = s;
    }
}

// ---------------- CSR scatter ----------------
__global__ __launch_bounds__(256) void scatter_kernel(const int* __restrict__ edges,
                                                      int* __restrict__ cursor,
                                                      int* __restrict__ csr) {
    const int e = blockIdx.x * 256 + threadIdx.x;   // < NE
    const int pos = atomicAdd(&cursor[edges[e]], 1);
    csr[pos] = e;
}

// ---------------- softmax numerator + row sums ----------------
__global__ __launch_bounds__(256) void edge_p_kernel(const float* __restrict__ wbuf,
                                                     const int* __restrict__ edges,
                                                     const unsigned* __restrict__ rmk,
                                                     float* __restrict__ pexp,
                                                     float* __restrict__ rowsum) {
    const int t = blockIdx.x * 256 + threadIdx.x;   // < NH*NE
    const int h = t / NE;
    const int e = t - h * NE;
    const int src = edges[e];
    const unsigned k = rmk[h * NN + src];
    const unsigned b = (k & 0x80000000u) ? (k & 0x7FFFFFFFu) : ~k;
    const float m = __uint_as_float(b);
    const float pv = __expf(wbuf[t] - m);
    pexp[t] = pv;
    atomicAdd(&rowsum[h * NN + src], pv);
}

// ---------------- aggregation + ELU + residual + LayerNorm ----------------
// one block per node row; wave w == head w; lane covers hd = lane, lane+32
__global__ __launch_bounds__(256) void agg_ln_kernel(const float* __restrict__ P,
                                                     const float* __restrict__ pexp,
                                                     const float* __restrict__ rowsum,
                                                     const int* __restrict__ rowstart,
                                                     const int* __restrict__ csr,
                                                     const int* __restrict__ edges,
                                                     const float* __restrict__ x,
                                                     const float* __restrict__ gamma,
                                                     const float* __restrict__ beta,
                                                     float* __restrict__ out) {
    const int i    = blockIdx.x;
    const int lane = threadIdx.x & 31;
    const int h    = threadIdx.x >> 5;
    const int s0 = rowstart[i], s1 = rowstart[i + 1];
    const float rs  = rowsum[h * NN + i];
    const float inv = (rs > 0.f) ? 1.f / rs : 0.f;   // empty row -> softmax NaN -> 0

    float a0 = 0.f, a1 = 0.f;
    for (int idx = s0; idx < s1; ++idx) {
        const int e = csr[idx];
        const float coef = pexp[h * NE + e] * inv;
        const int dst = edges[NE + e];
        const float* vr = P + dst * PCOLS + h * HD;   // value projection (cols 0..511, +bv)
        a0 = fmaf(coef, vr[lane],      a0);
        a1 = fmaf(coef, vr[lane + 32], a1);
    }

    const int c0 = h * HD + lane, c1 = c0 + 32;
    float y0 = (a0 > 0.f ? a0 : __expf(a0) - 1.f) + x[i * HID + c0];  // elu + residual
    float y1 = (a1 > 0.f ? a1 : __expf(a1) - 1.f) + x[i * HID + c1];

    float s = y0 + y1, ss = y0 * y0 + y1 * y1;
    #pragma unroll
    for (int o = 16; o > 0; o >>= 1) {
        s  += __shfl_xor(s,  o, 32);
        ss += __shfl_xor(ss, o, 32);
    }
    __shared__ float red[2][NH];
    if (lane == 0) { red[0][h] = s; red[1][h] = ss; }
    __syncthreads();
    float tot = 0.f, tot2 = 0.f;
    #pragma unroll
    for (int j = 0; j < NH; ++j) { tot += red[0][j]; tot2 += red[1][j]; }
    const float mu   = tot * (1.f / HID);
    const float var  = tot2 * (1.f / HID) - mu * mu;
    const float rstd = rsqrtf(var + LN_EPS);
    out[i * HID + c0] = (y0 - mu) * rstd * gamma[c0] + beta[c0];
    out[i * HID + c1] = (y1 - mu) * rstd * gamma[c1] + beta[c1];
}

// ---------------- host launcher ----------------
extern "C" void kernel_launch(void* const* d_in, const int* in_sizes, int n_in,
                              void* d_out, int out_size, void* d_ws, size_t ws_size,
                              hipStream_t stream) {
    const float* x     = (const float*)d_in[0];
    const int*   edges = (const int*)  d_in[1];
    const float* Wv    = (const float*)d_in[2];
    const float* bv    = (const float*)d_in[3];
    const float* Ww    = (const float*)d_in[4];
    const float* bw    = (const float*)d_in[5];
    const float* Wa    = (const float*)d_in[6];
    const float* ba    = (const float*)d_in[7];
    const float* gamma = (const float*)d_in[8];
    const float* beta  = (const float*)d_in[9];
    float* out = (float*)d_out;

    float*    ws      = (float*)d_ws;
    float*    P       = ws + OFF_P;
    float*    wbuf    = ws + OFF_W;
    float*    pexp    = ws + OFF_PEXP;
    unsigned* rmk     = (unsigned*)(ws + OFF_RMK);
    float*    rowsum  = ws + OFF_RSUM;
    int*      counts  = (int*)(ws + OFF_CNT);
    int*      rstart  = (int*)(ws + OFF_START);
    int*      cursor  = (int*)(ws + OFF_CUR);
    int*      csr     = (int*)(ws + OFF_CSR);

    init_kernel<<<(ZERO_WORDS + 255) / 256, 256, 0, stream>>>(rmk);

    dim3 ggrid(NN / 128, PCOLS / 128);
    gemm_kernel<<<ggrid, 256, 0, stream>>>(x, Wv, bv, Ww, bw, P);

    edge_w_kernel<<<(NH * NE) / 8, 256, 0, stream>>>(P, edges, Wa, ba, wbuf, rmk, counts);

    scan_kernel<<<1, 32, 0, stream>>>(counts, rstart, cursor);

    scatter_kernel<<<NE / 256, 256, 0, stream>>>(edges, cursor, csr);

    edge_p_kernel<<<(NH * NE) / 256, 256, 0, stream>>>(wbuf, edges, rmk, pexp, rowsum);

    agg_ln_kernel<<<NN, 256, 0, stream>>>(P, pexp, rowsum, rstart, csr, edges,
                                          x, gamma, beta, out);
}